// AttractionBranch_37082747634276
// MI455X (gfx1250) — compile-verified
//
#include <hip/hip_runtime.h>

typedef float v2f __attribute__((ext_vector_type(2)));
typedef float v8f __attribute__((ext_vector_type(8)));

#define DH 128

// ---------------------------------------------------------------------------
// Dense GEMM: C[M,128] = A[M,128] @ W[128,128] (+bias)(+relu), f32 via WMMA.
// mode: 0 = plain, 1 = bias+relu, 2 = bias only.
// One workgroup = 8 wave32 = 128 rows; W staged in 64KB LDS; each wave owns a
// 16-row strip x full 128 cols (8 C tiles, 64 accumulator VGPRs).
// A fragments are software-pipelined 2 k-steps ahead so ~16 in-flight WMMAs
// cover the global-load latency instead of stalling on s_wait_loadcnt 0.
// ---------------------------------------------------------------------------
__global__ __launch_bounds__(256) void gcn_gemm128(
    const float* __restrict__ A, const float* __restrict__ W,
    const float* __restrict__ bias, float* __restrict__ C,
    int M, int mode)
{
    __shared__ float Ws[DH * DH];  // 64 KB, well under 320 KB/WGP

    const int tid = threadIdx.x;
    {   // cooperative 128-bit staged copy of W into LDS
        const float4* src = (const float4*)W;
        float4*       dst = (float4*)Ws;
        for (int i = tid; i < DH * DH / 4; i += 256) dst[i] = src[i];
    }
    __syncthreads();

    const int wave = tid >> 5;
    const int lane = tid & 31;
    const int row0 = blockIdx.x * 128 + wave * 16;
    if (row0 >= M) return;  // wave-uniform: EXEC stays all-1s for WMMA

    const int mrow  = lane & 15;        // A-frag M index (ISA 16x4 f32 layout)
    const int khalf = (lane >> 4) * 2;  // lanes 0-15 -> K 0/1, 16-31 -> K 2/3
    const float* Arow = A + (size_t)(row0 + mrow) * DH;

    v8f acc[8] = {};  // 8 N-tiles of 16 -> 128 columns

    // 2-deep A-fragment prefetch pipeline
    float2 a0 = *(const float2*)(Arow + khalf);
    float2 a1 = *(const float2*)(Arow + 4 + khalf);

    #pragma unroll
    for (int k = 0; k < DH; k += 4) {
        v2f a; a.x = a0.x; a.y = a0.y;
        a0 = a1;
        if (k + 8 < DH)
            a1 = *(const float2*)(Arow + k + 8 + khalf);
        #pragma unroll
        for (int t = 0; t < 8; ++t) {
            const int ncol = t * 16 + (lane & 15);  // B-frag N index
            v2f b;
            b.x = Ws[(k + khalf) * DH + ncol];
            b.y = Ws[(k + khalf + 1) * DH + ncol];
            acc[t] = __builtin_amdgcn_wmma_f32_16x16x4_f32(
                false, a, false, b, (short)0, acc[t], false, false);
        }
    }

    // C/D layout: VGPR v, lane l -> M = v + 8*(l>>4), N = l&15
    const int rowoff = (lane >> 4) * 8;
    #pragma unroll
    for (int t = 0; t < 8; ++t) {
        const int c  = t * 16 + (lane & 15);
        const float bv = (mode != 0) ? bias[c] : 0.0f;
        #pragma unroll
        for (int v = 0; v < 8; ++v) {
            float val = acc[t][v] + bv;
            if (mode == 1) val = fmaxf(val, 0.0f);
            C[(size_t)(row0 + rowoff + v) * DH + c] = val;
        }
    }
}

// ---------------------------------------------------------------------------
// Degree / normalization (self-loop contributes deg init = 1).
// ---------------------------------------------------------------------------
__global__ void gcn_deg_init(float* deg, int N)
{
    int i = blockIdx.x * blockDim.x + threadIdx.x;
    if (i < N) deg[i] = 1.0f;
}

__global__ void gcn_deg_count(const int* __restrict__ col, float* deg, int E)
{
    int e = blockIdx.x * blockDim.x + threadIdx.x;
    if (e < E) atomicAdd(&deg[col[e]], 1.0f);
}

__global__ void gcn_deg_rsqrt(float* deg, int N)
{
    int i = blockIdx.x * blockDim.x + threadIdx.x;
    if (i < N) deg[i] = rsqrtf(fmaxf(deg[i], 1.0f));
}

// ---------------------------------------------------------------------------
// agg[i,:] = dinv[i]^2 * t[i,:]   (self-loop message; also initializes agg)
// One wave per node, one float4 per lane.
// ---------------------------------------------------------------------------
__global__ void gcn_selfloop(const float* __restrict__ dinv,
                             const float* __restrict__ t,
                             float* __restrict__ agg, int N)
{
    int g = blockIdx.x * blockDim.x + threadIdx.x;
    int i = g >> 5;
    int q = g & 31;
    if (i >= N) return;
    float s = dinv[i]; s *= s;
    float4 v = ((const float4*)(t + (size_t)i * DH))[q];
    v.x *= s; v.y *= s; v.z *= s; v.w *= s;
    ((float4*)(agg + (size_t)i * DH))[q] = v;
}

// ---------------------------------------------------------------------------
// agg[col,:] += dinv[row]*dinv[col] * t[row,:]   (segment_sum via f32 atomics)
// One wave per edge, one float4 per lane -> 4 no-return global f32 atomics.
// Working set (~77 MB) lives in the 192MB L2, so random traffic stays on-chip.
// ---------------------------------------------------------------------------
__global__ void gcn_scatter(const int* __restrict__ row,
                            const int* __restrict__ col,
                            const float* __restrict__ dinv,
                            const float* __restrict__ t,
                            float* __restrict__ agg, int E)
{
    int g = blockIdx.x * blockDim.x + threadIdx.x;
    int e = g >> 5;
    int q = g & 31;
    if (e >= E) return;
    const int r = row[e];
    const int c = col[e];
    const float w = dinv[r] * dinv[c];
    const float4 v = ((const float4*)(t + (size_t)r * DH))[q];
    float* dst = agg + (size_t)c * DH + q * 4;
    atomicAdd(dst + 0, w * v.x);
    atomicAdd(dst + 1, w * v.y);
    atomicAdd(dst + 2, w * v.z);
    atomicAdd(dst + 3, w * v.w);
}

// ---------------------------------------------------------------------------
// h[i,:] += relu(agg[i,:] + b[:])   (residual + bias + relu), float4 per thread
// ---------------------------------------------------------------------------
__global__ void gcn_combine(float* __restrict__ h,
                            const float* __restrict__ agg,
                            const float* __restrict__ bias, int N)
{
    int g = blockIdx.x * blockDim.x + threadIdx.x;
    if (g >= N * (DH / 4)) return;
    const int q = g & 31;
    const float4 a = ((const float4*)agg)[g];
    const float4 b = ((const float4*)bias)[q];
    float4 hh = ((float4*)h)[g];
    hh.x += fmaxf(a.x + b.x, 0.0f);
    hh.y += fmaxf(a.y + b.y, 0.0f);
    hh.z += fmaxf(a.z + b.z, 0.0f);
    hh.w += fmaxf(a.w + b.w, 0.0f);
    ((float4*)h)[g] = hh;
}

// ---------------------------------------------------------------------------
extern "C" void kernel_launch(void* const* d_in, const int* in_sizes, int n_in,
                              void* d_out, int out_size, void* d_ws, size_t ws_size,
                              hipStream_t stream)
{
    const float* x     = (const float*)d_in[0];
    const int*   eidx  = (const int*)d_in[1];   // [2,E]: rows then cols
    const float* W_in  = (const float*)d_in[2];
    const float* b_in  = (const float*)d_in[3];
    const float* Wl[3] = {(const float*)d_in[4], (const float*)d_in[6], (const float*)d_in[8]};
    const float* bl[3] = {(const float*)d_in[5], (const float*)d_in[7], (const float*)d_in[9]};
    const float* W_out = (const float*)d_in[10];
    const float* b_out = (const float*)d_in[11];

    const int N = in_sizes[0] / DH;  // 50000 (multiple of 16)
    const int E = in_sizes[1] / 2;   // 800000
    const int* row = eidx;
    const int* col = eidx + E;

    // workspace layout: h (N*128 f32) | agg (N*128 f32) | deg/dinv (N f32)
    float* h   = (float*)d_ws;
    float* agg = h + (size_t)N * DH;
    float* deg = agg + (size_t)N * DH;
    float* t   = (float*)d_out;  // reuse output buffer as per-layer h@W scratch

    const int B = 256;
    const int gemmBlocks = (N + 127) / 128;

    // normalization: deg = 1 (self-loop) + in-degree; dinv = rsqrt(deg)
    gcn_deg_init <<<(N + B - 1) / B, B, 0, stream>>>(deg, N);
    gcn_deg_count<<<(E + B - 1) / B, B, 0, stream>>>(col, deg, E);
    gcn_deg_rsqrt<<<(N + B - 1) / B, B, 0, stream>>>(deg, N);

    // h = relu(x @ W_in + b_in)
    gcn_gemm128<<<gemmBlocks, 256, 0, stream>>>(x, W_in, b_in, h, N, 1);

    for (int l = 0; l < 3; ++l) {
        gcn_gemm128 <<<gemmBlocks, 256, 0, stream>>>(h, Wl[l], nullptr, t, N, 0);
        gcn_selfloop<<<((N * 32) + B - 1) / B, B, 0, stream>>>(deg, t, agg, N);
        gcn_scatter <<<((E * 32) + B - 1) / B, B, 0, stream>>>(row, col, deg, t, agg, E);
        gcn_combine <<<((N * (DH / 4)) + B - 1) / B, B, 0, stream>>>(h, agg, bl[l], N);
    }

    // out = h @ W_out + b_out
    gcn_gemm128<<<gemmBlocks, 256, 0, stream>>>(h, W_out, b_out, (float*)d_out, N, 2);
}